// ModelNew_25056839205117
// MI455X (gfx1250) — compile-verified
//
#include <hip/hip_runtime.h>
#include <hip/hip_bf16.h>

// SK-Net selective-kernel fuse on MI455X (gfx1250, wave32).
//
// Roofline: ~514 MB mandatory HBM traffic, ~0.26 GFLOP -> purely bandwidth
// bound (~22 us floor @ 23.3 TB/s). Data path chosen for CDNA5:
//   * GLOBAL_LOAD_ASYNC_TO_LDS_B128 staging of all 4 input planes (ASYNCcnt,
//     no VGPR occupancy while in flight), s_wait_asynccnt + barrier,
//     then DS_LOAD_B128 reads.
//   * nontemporal GLOBAL_STORE_B128 for out (stream-once data, keep L2 clean)
//   * blockIdx-uniform softmax -> scalar-pipe s_load/s_fmac
//   * HW-sum reduction: V_WMMA_F32_16X16X4_F32 with B=ones collapses the
//     32-lane partials on the matrix pipe (exact f32), + one shfl_xor(16),
//     then an 8-entry LDS combine across the block's 8 wave32s.

#define BB 32
#define CC 256
#define HW 3136        // 56*56
#define HW4 784        // HW/4 float4s per plane
#define BC (BB * CC)   // 8192 planes
#define PLANE_ELEMS ((size_t)BC * (size_t)HW)  // 25,690,112

typedef float v2f __attribute__((ext_vector_type(2)));
typedef float v4f __attribute__((ext_vector_type(4)));
typedef float v8f __attribute__((ext_vector_type(8)));
typedef int   v4i __attribute__((ext_vector_type(4)));

#if defined(__gfx1250__) && __has_builtin(__builtin_amdgcn_global_load_async_to_lds_b128)
#define USE_ASYNC 1
#else
#define USE_ASYNC 0
#endif

#if defined(__gfx1250__) && __has_builtin(__builtin_amdgcn_wmma_f32_16x16x4_f32)
#define USE_WMMA_RED 1
#else
#define USE_WMMA_RED 0
#endif

__global__ __launch_bounds__(256) void sk_fuse_kernel(
    const float* __restrict__ x0, const float* __restrict__ x1,
    const float* __restrict__ x2, const float* __restrict__ x3,
    const float* __restrict__ attn,     // [K=4, B*C]
    float* __restrict__ out,            // [B, C, H, W]
    float* __restrict__ s_out,          // [B, C]
    float* __restrict__ usum_out)       // [B, C]
{
    const int p   = blockIdx.x;         // plane id = b*C + c (uniform)
    const int tid = threadIdx.x;

    // ---- per-plane softmax weights (uniform -> scalar pipe) ----
    const float l0 = attn[p];
    const float l1 = attn[p + BC];
    const float l2 = attn[p + 2 * BC];
    const float l3 = attn[p + 3 * BC];
    float m = fmaxf(fmaxf(l0, l1), fmaxf(l2, l3));
    float e0 = expf(l0 - m), e1 = expf(l1 - m), e2 = expf(l2 - m), e3 = expf(l3 - m);
    float inv = 1.0f / (e0 + e1 + e2 + e3);
    const float w0 = e0 * inv, w1 = e1 * inv, w2 = e2 * inv, w3 = e3 * inv;

    const size_t base = (size_t)p * (size_t)HW4;   // in float4 units
    const v4f* __restrict__ p0 = (const v4f*)x0 + base;
    const v4f* __restrict__ p1 = (const v4f*)x1 + base;
    const v4f* __restrict__ p2 = (const v4f*)x2 + base;
    const v4f* __restrict__ p3 = (const v4f*)x3 + base;
    v4f* __restrict__ po = (v4f*)out + base;

    float acc = 0.0f;   // per-thread partial of sum_{hw} u

#if USE_ASYNC
    // ---- async-stage all 4 branch planes into LDS (50,176 B) ----
    __shared__ v4f tile[4][HW4];
    typedef __attribute__((address_space(1))) v4i* gbl_v4i;
    typedef __attribute__((address_space(3))) v4i* lds_v4i;

    for (int i = tid; i < HW4; i += 256) {
        __builtin_amdgcn_global_load_async_to_lds_b128((gbl_v4i)(p0 + i), (lds_v4i)&tile[0][i], 0, 0);
        __builtin_amdgcn_global_load_async_to_lds_b128((gbl_v4i)(p1 + i), (lds_v4i)&tile[1][i], 0, 0);
        __builtin_amdgcn_global_load_async_to_lds_b128((gbl_v4i)(p2 + i), (lds_v4i)&tile[2][i], 0, 0);
        __builtin_amdgcn_global_load_async_to_lds_b128((gbl_v4i)(p3 + i), (lds_v4i)&tile[3][i], 0, 0);
    }
#if __has_builtin(__builtin_amdgcn_s_wait_asynccnt)
    __builtin_amdgcn_s_wait_asynccnt(0);
#else
    asm volatile("s_wait_asynccnt 0x0" ::: "memory");
#endif
    __syncthreads();

    for (int i = tid; i < HW4; i += 256) {
        v4f a0 = tile[0][i];
        v4f a1 = tile[1][i];
        v4f a2 = tile[2][i];
        v4f a3 = tile[3][i];

        v4f u = a0 + a1 + a2 + a3;
        acc += u.x + u.y + u.z + u.w;

        v4f o = a0 * w0 + a1 * w1 + a2 * w2 + a3 * w3;
        __builtin_nontemporal_store(o, po + i);
    }
#else
    // ---- fallback: direct NT b128 streaming (round-1 path) ----
    for (int i = tid; i < HW4; i += 256) {
        if (i + 256 < HW4) {
            __builtin_prefetch(p0 + i + 256, 0, 0);
            __builtin_prefetch(p1 + i + 256, 0, 0);
            __builtin_prefetch(p2 + i + 256, 0, 0);
            __builtin_prefetch(p3 + i + 256, 0, 0);
        }
        v4f a0 = __builtin_nontemporal_load(p0 + i);
        v4f a1 = __builtin_nontemporal_load(p1 + i);
        v4f a2 = __builtin_nontemporal_load(p2 + i);
        v4f a3 = __builtin_nontemporal_load(p3 + i);

        v4f u = a0 + a1 + a2 + a3;
        acc += u.x + u.y + u.z + u.w;

        v4f o = a0 * w0 + a1 * w1 + a2 * w2 + a3 * w3;
        __builtin_nontemporal_store(o, po + i);
    }
#endif

    // ---- intra-wave reduction of acc (wave32, EXEC all-ones here) ----
#if USE_WMMA_RED
    {
        // A[row=lane%16][K]: lanes 0-15 supply (K0,K1)=(acc,0),
        // lanes 16-31 supply (K2,K3)=(acc,0). B = ones(4x16).
        // => D[i][j] = acc_i + acc_{i+16}; each lane sums its 8 D rows,
        //    one shfl_xor(16) completes the 32-lane sum. Exact f32.
        v2f a; a.x = acc; a.y = 0.0f;
        v2f b; b.x = 1.0f; b.y = 1.0f;
        v8f c = {};
        c = __builtin_amdgcn_wmma_f32_16x16x4_f32(
            /*neg_a=*/false, a, /*neg_b=*/false, b,
            /*c_mod=*/(short)0, c, /*reuse_a=*/false, /*reuse_b=*/false);
        float s8 = 0.0f;
        #pragma unroll
        for (int r = 0; r < 8; ++r) s8 += c[r];
        acc = s8 + __shfl_xor(s8, 16, 32);
    }
#else
    #pragma unroll
    for (int off = 16; off > 0; off >>= 1)
        acc += __shfl_xor(acc, off, 32);
#endif

    // ---- cross-wave combine (8 wave32s per block) ----
    __shared__ float wave_sums[8];
    const int lane = tid & 31;
    const int wave = tid >> 5;
    if (lane == 0) wave_sums[wave] = acc;
    __syncthreads();

    if (tid == 0) {
        float t = 0.0f;
        #pragma unroll
        for (int j = 0; j < 8; ++j) t += wave_sums[j];
        usum_out[p] = t;
        s_out[p]    = t * (1.0f / (float)HW);
    }
}

extern "C" void kernel_launch(void* const* d_in, const int* in_sizes, int n_in,
                              void* d_out, int out_size, void* d_ws, size_t ws_size,
                              hipStream_t stream) {
    const float* x0   = (const float*)d_in[0];
    const float* x1   = (const float*)d_in[1];
    const float* x2   = (const float*)d_in[2];
    const float* x3   = (const float*)d_in[3];
    const float* attn = (const float*)d_in[4];

    float* out  = (float*)d_out;              // [B,C,H,W] flat
    float* s    = out + PLANE_ELEMS;          // [B,C]
    float* usum = s + BC;                     // [B,C]

    sk_fuse_kernel<<<dim3(BC), dim3(256), 0, stream>>>(x0, x1, x2, x3, attn,
                                                       out, s, usum);
}